// Encoding_73332271612207
// MI455X (gfx1250) — compile-verified
//
#include <hip/hip_runtime.h>
#include <math.h>

// Problem constants (match reference)
#define KC 32      // clusters
#define DD 512     // feature dim
#define NB 16      // batch
#define NN 4096    // tokens per batch (H*W)
#define NT 256     // tokens per block (kernels A and B)
#define NSPLIT 16  // NN / NT

typedef __attribute__((ext_vector_type(2))) float v2f;  // WMMA f32 A/B fragment (16x4 / 4x16)
typedef __attribute__((ext_vector_type(8))) float v8f;  // WMMA f32 C/D fragment (16x16)

// ---- workspace layout (in floats) ----
#define AMAT_OFF  ((size_t)0)
#define AMAT_SZ   ((size_t)NB * KC * NN)              // 2,097,152 floats (8 MB)  A[b][k][n]
#define SAP_OFF   (AMAT_OFF + AMAT_SZ)
#define SAP_SZ    ((size_t)NB * NSPLIT * KC)          // 8,192 floats            sa partials
#define C2_OFF    (SAP_OFF + SAP_SZ)                  // ||c_k||^2
#define S2_OFF    (C2_OFF + KC)                       // scale_k^2
#define EPART_OFF (S2_OFF + KC)
#define EPART_SZ  ((size_t)NB * NSPLIT * KC * DD)     // 4,194,304 floats (16 MB) e partials
// total ws use: ~25.2 MB

// ---------------- kernel 0: per-cluster constants ----------------
__global__ void k0_prep(const float* __restrict__ codes, const float* __restrict__ scale,
                        float* __restrict__ c2, float* __restrict__ s2) {
    int k = threadIdx.x;
    if (k < KC) {
        float s = 0.f;
        for (int d = 0; d < DD; ++d) { float c = codes[k * DD + d]; s += c * c; }
        c2[k] = s;
        float sc = scale[k];
        s2[k] = sc * sc;
    }
}

// ---------------- kernel A: xc GEMM + softmax -> A weights ----------------
// grid: NB * (NN/NT) blocks; block = 256 threads (8 wave32)
// Each block handles one (batch, 256-token chunk).
//   xc[k,n] = sum_d codes[k,d] * x[b,d,n]   (x memory layout is (B, D, N): n contiguous!)
// Wave w owns tokens [32w, 32w+32): 2 (k-tiles) x 2 (n-tiles) of f32 WMMA accumulators.
__global__ __launch_bounds__(256)
void kA_assign(const float* __restrict__ x, const float* __restrict__ codes,
               const float* __restrict__ c2g, const float* __restrict__ s2g,
               float* __restrict__ Amat, float* __restrict__ sap) {
    __shared__ float cds[KC * 516];     // codes, rows padded 512->516 (bank spread, 8B align)
    __shared__ float xtile[16 * 260];   // x panel: 16 d-rows x 256 tokens, pad->260
    __shared__ float xcbuf[NT * 33];    // xc then a, [token][k], pad 33
    __shared__ float x2buf[4 * NT];     // 4-way partial ||x||^2 per token (deterministic reduce)
    __shared__ float c2s[KC], s2s[KC];

    const int tid  = threadIdx.x;
    const int wave = tid >> 5;
    const int lane = tid & 31;
    const int l15  = lane & 15;
    const int lh   = lane >> 4;

    const int b     = blockIdx.x >> 4;      // 16 chunks per batch
    const int chunk = blockIdx.x & 15;
    const int n0    = chunk * NT;
    const float* xb = x + (size_t)b * DD * NN;

    // preload codes (32 x 512) into LDS
    for (int i = tid; i < KC * (DD / 4); i += 256) {   // 16 iters
        int row = i >> 7, c4 = i & 127;
        float4 v = *(const float4*)(codes + row * DD + c4 * 4);
        *(float4*)(&cds[row * 516 + c4 * 4]) = v;
    }
    if (tid < KC) { c2s[tid] = c2g[tid]; s2s[tid] = s2g[tid]; }

    v8f acc00 = {}, acc01 = {}, acc10 = {}, acc11 = {};
    float x2p[4] = {0.f, 0.f, 0.f, 0.f};
    const int tw = wave * 32;

    for (int dp = 0; dp < DD; dp += 16) {
        __syncthreads();
        // stage x panel: rows are d (contiguous tokens) -> fully coalesced b128 loads
        for (int i = tid; i < 16 * 64; i += 256) {     // 4 iters; column set fixed per thread
            int row = i >> 6, c4 = i & 63;
            float4 v = *(const float4*)(xb + (size_t)(dp + row) * NN + n0 + c4 * 4);
            *(float4*)(&xtile[row * 260 + c4 * 4]) = v;
            x2p[0] += v.x * v.x; x2p[1] += v.y * v.y;
            x2p[2] += v.z * v.z; x2p[3] += v.w * v.w;
        }
        __syncthreads();
        #pragma unroll
        for (int d4 = 0; d4 < 16; d4 += 4) {
            // A (codes 16x4): K-idx = 2*lh + v, M = l15
            int ccol = dp + d4 + 2 * lh;
            v2f a0 = *(const v2f*)(&cds[l15 * 516 + ccol]);
            v2f a1 = *(const v2f*)(&cds[(16 + l15) * 516 + ccol]);
            // B (x 4x16): row d = d4 + 2*lh + v, col = token
            int xr = (d4 + 2 * lh) * 260;
            v2f b0, b1;
            b0.x = xtile[xr + tw + l15];        b0.y = xtile[xr + 260 + tw + l15];
            b1.x = xtile[xr + tw + 16 + l15];   b1.y = xtile[xr + 260 + tw + 16 + l15];
            acc00 = __builtin_amdgcn_wmma_f32_16x16x4_f32(false, a0, false, b0, (short)0, acc00, false, false);
            acc01 = __builtin_amdgcn_wmma_f32_16x16x4_f32(false, a0, false, b1, (short)0, acc01, false, false);
            acc10 = __builtin_amdgcn_wmma_f32_16x16x4_f32(false, a1, false, b0, (short)0, acc10, false, false);
            acc11 = __builtin_amdgcn_wmma_f32_16x16x4_f32(false, a1, false, b1, (short)0, acc11, false, false);
        }
    }

    // dump xc tiles: C layout row k = j + 8*lh, col = l15
    #pragma unroll
    for (int j = 0; j < 8; ++j) {
        int k0 = j + 8 * lh;
        xcbuf[(tw + l15)      * 33 + k0     ] = acc00[j];
        xcbuf[(tw + l15)      * 33 + 16 + k0] = acc10[j];
        xcbuf[(tw + 16 + l15) * 33 + k0     ] = acc01[j];
        xcbuf[(tw + 16 + l15) * 33 + 16 + k0] = acc11[j];
    }
    { // per-thread ||x||^2 partials (columns owned: (tid&63)*4 + j, 4 row-groups)
        int m = tid >> 6, cbase = (tid & 63) * 4;
        #pragma unroll
        for (int j = 0; j < 4; ++j) x2buf[m * NT + cbase + j] = x2p[j];
    }
    __syncthreads();

    // softmax over K=32 for token t (note: softmax of +dist, as in reference)
    {
        int t = tid;
        float x2 = x2buf[t] + x2buf[NT + t] + x2buf[2 * NT + t] + x2buf[3 * NT + t];
        float dst[KC];
        float mx = -3.4e38f;
        #pragma unroll
        for (int k = 0; k < KC; ++k) {
            float v = s2s[k] * (x2 - 2.f * xcbuf[t * 33 + k] + c2s[k]);
            dst[k] = v;
            mx = fmaxf(mx, v);
        }
        float sum = 0.f;
        #pragma unroll
        for (int k = 0; k < KC; ++k) { float e = __expf(dst[k] - mx); dst[k] = e; sum += e; }
        float inv = 1.f / sum;
        #pragma unroll
        for (int k = 0; k < KC; ++k) {
            float a = dst[k] * inv;
            Amat[((size_t)b * KC + k) * NN + n0 + t] = a;   // coalesced across t
            xcbuf[t * 33 + k] = a;
        }
    }
    __syncthreads();
    // deterministic column sums: sa_partial[b][chunk][k]
    if (tid < KC) {
        float s = 0.f;
        for (int t = 0; t < NT; ++t) s += xcbuf[t * 33 + tid];
        sap[((size_t)b * NSPLIT + chunk) * KC + tid] = s;
    }
}

// ---------------- kernel B: e1 partial = A(K x n-range) * X^T ----------------
// grid: NB * NSPLIT; block 256 threads. Wave w owns d-slice [64w, 64w+64):
// 2 k-tiles x 4 d-tiles of accumulators. Inner dim = n (256 tokens, paneled by 32).
// x re-read is L2-resident (128 MB < 192 MB L2).
__global__ __launch_bounds__(256)
void kB_aggregate(const float* __restrict__ x, const float* __restrict__ Amat,
                  float* __restrict__ epart) {
    __shared__ float xnp[DD * 36];      // 512 d-rows x 32 tokens, pad 32->36 (bank spread)
    __shared__ float achunk[KC * 260];  // A weights: 32 k x 256 tokens

    const int tid  = threadIdx.x;
    const int wave = tid >> 5;
    const int lane = tid & 31;
    const int l15  = lane & 15;
    const int lh   = lane >> 4;

    const int b  = blockIdx.x >> 4;
    const int s  = blockIdx.x & 15;
    const int r0 = s * NT;
    const float* xb = x + (size_t)b * DD * NN;

    for (int i = tid; i < KC * 64; i += 256) {         // 8 iters, coalesced
        int row = i >> 6, c4 = i & 63;
        float4 v = *(const float4*)(Amat + ((size_t)b * KC + row) * NN + r0 + c4 * 4);
        *(float4*)(&achunk[row * 260 + c4 * 4]) = v;
    }

    v8f acc[2][4] = {};

    for (int np = 0; np < NT; np += 32) {
        __syncthreads();
        // stage x panel: each row = 128 contiguous bytes from global
        for (int i = tid; i < DD * 8; i += 256) {      // 16 iters
            int row = i >> 3, c4 = i & 7;
            float4 v = *(const float4*)(xb + (size_t)row * NN + r0 + np + c4 * 4);
            *(float4*)(&xnp[row * 36 + c4 * 4]) = v;
        }
        __syncthreads();
        #pragma unroll 4
        for (int n4 = 0; n4 < 32; n4 += 4) {
            // A' (weights 16x4): row k = l15 (+16), K-idx n = n4 + 2*lh + v
            int an = np + n4 + 2 * lh;
            v2f a0 = *(const v2f*)(&achunk[l15 * 260 + an]);
            v2f a1 = *(const v2f*)(&achunk[(16 + l15) * 260 + an]);
            #pragma unroll
            for (int dt = 0; dt < 4; ++dt) {
                // B' (x 4x16): row n = n4 + 2*lh + v, col d = 64*wave + 16*dt + l15
                int drow = wave * 64 + dt * 16 + l15;
                v2f bb = *(const v2f*)(&xnp[drow * 36 + n4 + 2 * lh]);
                acc[0][dt] = __builtin_amdgcn_wmma_f32_16x16x4_f32(false, a0, false, bb, (short)0, acc[0][dt], false, false);
                acc[1][dt] = __builtin_amdgcn_wmma_f32_16x16x4_f32(false, a1, false, bb, (short)0, acc[1][dt], false, false);
            }
        }
    }

    float* ep = epart + ((size_t)b * NSPLIT + s) * KC * DD;
    #pragma unroll
    for (int kt = 0; kt < 2; ++kt)
        #pragma unroll
        for (int dt = 0; dt < 4; ++dt)
            #pragma unroll
            for (int j = 0; j < 8; ++j) {
                int k = kt * 16 + j + 8 * lh;
                int d = wave * 64 + dt * 16 + l15;
                ep[(size_t)k * DD + d] = acc[kt][dt][j];
            }
}

// ---------------- kernel C: reduce partials, subtract sa*codes ----------------
__global__ __launch_bounds__(256)
void kC_final(const float* __restrict__ epart, const float* __restrict__ sap,
              const float* __restrict__ codes, float* __restrict__ out) {
    size_t idx = (size_t)blockIdx.x * 256 + threadIdx.x;  // over B*K*D = 262144
    int d = (int)(idx & (DD - 1));
    int k = (int)((idx >> 9) & (KC - 1));
    int b = (int)(idx >> 14);
    float acc = 0.f;
    #pragma unroll
    for (int s = 0; s < NSPLIT; ++s)
        acc += epart[(((size_t)b * NSPLIT + s) * KC + k) * DD + d];
    float sa = 0.f;
    #pragma unroll
    for (int c = 0; c < NSPLIT; ++c)
        sa += sap[((size_t)b * NSPLIT + c) * KC + k];
    out[idx] = acc - sa * codes[k * DD + d];
}

extern "C" void kernel_launch(void* const* d_in, const int* in_sizes, int n_in,
                              void* d_out, int out_size, void* d_ws, size_t ws_size,
                              hipStream_t stream) {
    const float* x     = (const float*)d_in[0];   // (B, D, H, W) = (B, D, N)
    const float* codes = (const float*)d_in[1];   // (K, D)
    const float* scale = (const float*)d_in[2];   // (K,)
    float* out = (float*)d_out;                   // (B, K, D)
    float* ws  = (float*)d_ws;

    float* Amat = ws + AMAT_OFF;
    float* sap  = ws + SAP_OFF;
    float* c2   = ws + C2_OFF;
    float* s2   = ws + S2_OFF;
    float* ep   = ws + EPART_OFF;

    k0_prep<<<1, 32, 0, stream>>>(codes, scale, c2, s2);
    kA_assign<<<NB * (NN / NT), 256, 0, stream>>>(x, codes, c2, s2, Amat, sap);
    kB_aggregate<<<NB * NSPLIT, 256, 0, stream>>>(x, Amat, ep);
    kC_final<<<(NB * KC * DD) / 256, 256, 0, stream>>>(ep, sap, codes, out);
}